// GAT_197568496078
// MI455X (gfx1250) — compile-verified
//
#include <hip/hip_runtime.h>
#include <hip/hip_bf16.h>
#include <math.h>

// ---------------------------------------------------------------------------
// GAT (2-layer, PyG GATConv semantics) for gfx1250 / MI455X.
// fp32 end-to-end; GEMMs use V_WMMA_F32_16X16X4_F32 (exact fp32 WMMA path).
// ---------------------------------------------------------------------------

typedef __attribute__((ext_vector_type(2))) float v2f;
typedef __attribute__((ext_vector_type(8))) float v8f;

__device__ __forceinline__ v8f wmma_f32_k4(v2f a, v2f b, v8f c) {
  // 8-arg pattern: (neg_a, A, neg_b, B, c_mod, C, reuse_a, reuse_b)
  return __builtin_amdgcn_wmma_f32_16x16x4_f32(
      false, a, false, b, (short)0, c, false, false);
}

// ------------------------------ GEMM ---------------------------------------
// out[M,128] = act(A[M,128]) @ W[128,128]
// If fuse_bias_relu != 0, act(v) = relu(v + bias[k]) applied to the A operand
// (this folds the layer-1 "aggregate + b1 -> relu" into the layer-2 GEMM).
__global__ __launch_bounds__(256) void gat_gemm128_wmma(
    const float* __restrict__ A, const float* __restrict__ W,
    const float* __restrict__ bias, float* __restrict__ out,
    int M, int fuse_bias_relu) {
  __shared__ float lW[128 * 128];  // 64 KB of the 320 KB WGP LDS
  const int t = threadIdx.x;
  for (int i = t * 4; i < 128 * 128; i += 256 * 4)
    *(float4*)&lW[i] = *(const float4*)&W[i];
  __syncthreads();

  const int wave = t >> 5;
  const int lane = t & 31;
  const int half = lane >> 4;  // 0: K pair {k0,k0+1}, 1: {k0+2,k0+3}
  const int l16  = lane & 15;
  const int rowbase = blockIdx.x * 128 + wave * 16;
  const int row  = rowbase + l16;
  const int rowc = row < M ? row : (M - 1);  // clamp loads; stores guarded

  // Preload A operands for all 32 k-steps (16x4 f32 A layout, ISA 7.12.2).
  v2f areg[32];
  const float* arow = A + (size_t)rowc * 128;
#pragma unroll
  for (int s = 0; s < 32; ++s) {
    const int k = s * 4 + half * 2;
    float2 v = *(const float2*)(arow + k);
    float ax = v.x, ay = v.y;
    if (fuse_bias_relu) {
      ax += bias[k];     ax = ax > 0.f ? ax : 0.f;
      ay += bias[k + 1]; ay = ay > 0.f ? ay : 0.f;
    }
    v2f a; a.x = ax; a.y = ay;
    areg[s] = a;
  }

  for (int j = 0; j < 8; ++j) {  // 8 column tiles of 16
    v8f c = {};
#pragma unroll
    for (int s = 0; s < 32; ++s) {
      const int k = s * 4 + half * 2;
      v2f b;
      b.x = lW[k * 128 + j * 16 + l16];
      b.y = lW[(k + 1) * 128 + j * 16 + l16];
      c = wmma_f32_k4(areg[s], b, c);
    }
    // C layout: VGPR r -> M = r (lanes 0-15) / r+8 (lanes 16-31), N = l16.
#pragma unroll
    for (int r = 0; r < 8; ++r) {
      const int orow = rowbase + r + half * 8;
      if (orow < M) out[(size_t)orow * 128 + j * 16 + l16] = c[r];
    }
  }
}

// --------------------- per-node attention coefficients ---------------------
// as[n,h] = <h[n, h*64 : h*64+64], att_src[h]> ; same for ad.
__global__ void gat_attn_coef(const float* __restrict__ h,
                              const float* __restrict__ att_src,
                              const float* __restrict__ att_dst,
                              float* __restrict__ as_, float* __restrict__ ad_,
                              int n) {
  const int i = blockIdx.x * blockDim.x + threadIdx.x;
  if (i >= n * 2) return;
  const int node = i >> 1, hd = i & 1;
  const float* hp = h + (size_t)node * 128 + hd * 64;
  const float* sp = att_src + hd * 64;
  const float* dp = att_dst + hd * 64;
  float accs = 0.f, accd = 0.f;
#pragma unroll 16
  for (int k = 0; k < 64; ++k) {
    const float v = hp[k];
    accs += v * sp[k];
    accd += v * dp[k];
  }
  as_[i] = accs;
  ad_[i] = accd;
}

// ------------------------------ utilities ----------------------------------
__global__ void gat_fill_f32(float* __restrict__ p, float v, int n) {
  const int i = blockIdx.x * blockDim.x + threadIdx.x;
  if (i < n) p[i] = v;
}

__device__ __forceinline__ void atomicMaxF(float* addr, float v) {
  if (v >= 0.f) atomicMax((int*)addr, __float_as_int(v));
  else          atomicMin((unsigned int*)addr, __float_as_uint(v));
}

__device__ __forceinline__ void edge_sd(const long long* __restrict__ ei,
                                        int e, int E, int* s, int* d) {
  if (e < E) { *s = (int)ei[e]; *d = (int)ei[(size_t)E + e]; }
  else       { *s = *d = e - E; }  // appended self-loops
}

// -------------------------- edge pass 1: segment max ------------------------
__global__ void gat_edge_max(const long long* __restrict__ ei,
                             const float* __restrict__ as_,
                             const float* __restrict__ ad_,
                             float* __restrict__ m, int E, int n) {
  const int e = blockIdx.x * blockDim.x + threadIdx.x;
  if (e >= E + n) return;
  int s, d; edge_sd(ei, e, E, &s, &d);
#pragma unroll
  for (int h = 0; h < 2; ++h) {
    float x = as_[s * 2 + h] + ad_[d * 2 + h];
    x = x > 0.f ? x : 0.2f * x;  // leaky_relu(0.2)
    atomicMaxF(&m[d * 2 + h], x);
  }
}

// ------------------------ edge pass 2: exp + seg-sum ------------------------
__global__ void gat_edge_expsum(const long long* __restrict__ ei,
                                const float* __restrict__ as_,
                                const float* __restrict__ ad_,
                                const float* __restrict__ m,
                                float* __restrict__ exb,
                                float* __restrict__ denom, int E, int n) {
  const int e = blockIdx.x * blockDim.x + threadIdx.x;
  if (e >= E + n) return;
  int s, d; edge_sd(ei, e, E, &s, &d);
#pragma unroll
  for (int h = 0; h < 2; ++h) {
    float x = as_[s * 2 + h] + ad_[d * 2 + h];
    x = x > 0.f ? x : 0.2f * x;
    const float ex = expf(x - m[d * 2 + h]);
    exb[(size_t)e * 2 + h] = ex;
    atomicAdd(&denom[d * 2 + h], ex);
  }
}

// -------------------- edge pass 3: weighted scatter-add ---------------------
// 32 lanes per edge, 4 features (one float4) per lane -> 128 features.
__global__ void gat_edge_agg(const long long* __restrict__ ei,
                             const float* __restrict__ feat,
                             const float* __restrict__ exb,
                             const float* __restrict__ denom,
                             float* __restrict__ agg, int E, int n) {
  const long long gid = (long long)blockIdx.x * blockDim.x + threadIdx.x;
  const int e = (int)(gid >> 5);
  if (e >= E + n) return;
  const int lane = (int)(gid & 31);
  int s, d; edge_sd(ei, e, E, &s, &d);
  const int f0 = lane * 4;
  const int hd = f0 >> 6;
  const float alpha = exb[(size_t)e * 2 + hd] / denom[d * 2 + hd];
  const float4 v = *(const float4*)&feat[(size_t)s * 128 + f0];
  float* o = &agg[(size_t)d * 128 + f0];
  atomicAdd(o + 0, v.x * alpha);
  atomicAdd(o + 1, v.y * alpha);
  atomicAdd(o + 2, v.z * alpha);
  atomicAdd(o + 3, v.w * alpha);
}

// ----------------------- final: mean over heads + bias ----------------------
__global__ void gat_finalize(const float* __restrict__ agg,
                             const float* __restrict__ b2,
                             float* __restrict__ out, int n) {
  const int i = blockIdx.x * blockDim.x + threadIdx.x;
  if (i >= n * 64) return;
  const int node = i >> 6, d = i & 63;
  out[i] = 0.5f * (agg[(size_t)node * 128 + d] + agg[(size_t)node * 128 + 64 + d])
         + b2[d];
}

// ---------------------------------------------------------------------------
static inline int cdiv(long long a, long long b) { return (int)((a + b - 1) / b); }

extern "C" void kernel_launch(void* const* d_in, const int* in_sizes, int n_in,
                              void* d_out, int out_size, void* d_ws, size_t ws_size,
                              hipStream_t stream) {
  const float*     x    = (const float*)d_in[0];
  const long long* ei   = (const long long*)d_in[1];  // jnp.int64 edge_index [2,E]
  const float*     W1   = (const float*)d_in[2];
  const float*     asr1 = (const float*)d_in[3];
  const float*     adt1 = (const float*)d_in[4];
  const float*     b1   = (const float*)d_in[5];
  const float*     W2   = (const float*)d_in[6];
  const float*     asr2 = (const float*)d_in[7];
  const float*     adt2 = (const float*)d_in[8];
  const float*     b2   = (const float*)d_in[9];
  float* out = (float*)d_out;

  const int N = in_sizes[0] / 128;
  const int E = in_sizes[1] / 2;
  const int Ep = E + N;

  // workspace layout (floats)
  float* bufA = (float*)d_ws;                    // N*128 (h features)
  float* bufB = bufA + (size_t)N * 128;          // N*128 (aggregate / gemm2 in)
  float* as_  = bufB + (size_t)N * 128;          // N*2
  float* ad_  = as_ + (size_t)N * 2;             // N*2
  float* m_   = ad_ + (size_t)N * 2;             // N*2
  float* den_ = m_  + (size_t)N * 2;             // N*2
  float* exb  = den_ + (size_t)N * 2;            // Ep*2

  const int TB = 256;
  const int gGemm = cdiv(N, 128);
  const int gN2   = cdiv((long long)N * 2, TB);
  const int gE    = cdiv(Ep, TB);
  const int gE32  = cdiv((long long)Ep * 32, TB);
  const float NEG_INF = -__builtin_huge_valf();

  // ---------------- layer 1 ----------------
  gat_gemm128_wmma<<<gGemm, TB, 0, stream>>>(x, W1, nullptr, bufA, N, 0);
  gat_attn_coef<<<gN2, TB, 0, stream>>>(bufA, asr1, adt1, as_, ad_, N);
  gat_fill_f32<<<gN2, TB, 0, stream>>>(m_, NEG_INF, N * 2);
  hipMemsetAsync(den_, 0, (size_t)N * 2 * sizeof(float), stream);
  hipMemsetAsync(bufB, 0, (size_t)N * 128 * sizeof(float), stream);
  gat_edge_max<<<gE, TB, 0, stream>>>(ei, as_, ad_, m_, E, N);
  gat_edge_expsum<<<gE, TB, 0, stream>>>(ei, as_, ad_, m_, exb, den_, E, N);
  gat_edge_agg<<<gE32, TB, 0, stream>>>(ei, bufA, exb, den_, bufB, E, N);

  // ---------------- layer 2 (bias+relu fused into GEMM A-load) ------------
  gat_gemm128_wmma<<<gGemm, TB, 0, stream>>>(bufB, W2, b1, bufA, N, 1);
  gat_attn_coef<<<gN2, TB, 0, stream>>>(bufA, asr2, adt2, as_, ad_, N);
  gat_fill_f32<<<gN2, TB, 0, stream>>>(m_, NEG_INF, N * 2);
  hipMemsetAsync(den_, 0, (size_t)N * 2 * sizeof(float), stream);
  hipMemsetAsync(bufB, 0, (size_t)N * 128 * sizeof(float), stream);
  gat_edge_max<<<gE, TB, 0, stream>>>(ei, as_, ad_, m_, E, N);
  gat_edge_expsum<<<gE, TB, 0, stream>>>(ei, as_, ad_, m_, exb, den_, E, N);
  gat_edge_agg<<<gE32, TB, 0, stream>>>(ei, bufA, exb, den_, bufB, E, N);

  gat_finalize<<<cdiv((long long)N * 64, TB), TB, 0, stream>>>(bufB, b2, out, N);
}